// EpsModel_16827681866268
// MI455X (gfx1250) — compile-verified
//
#include <hip/hip_runtime.h>
#include <cstdint>

// ============================================================================
// EpsModel: t2 = x@A0inv ; t4 = t2@t2 ; ft = A0inv @ (I - t2) @ (I + t4)
// B = 8,388,608 independent 2x2 fp32 matrices.
//
// Roofline (MI455X): 128 MiB in + 128 MiB out = 268 MB -> ~11.5 us @ 23.3 TB/s.
// ~70 FLOPs/matrix -> ~0.6 GFLOP total, AI ~2.2 FLOP/B: strictly HBM-bound.
// WMMA mapping would be <=12.5%-utilized (K=2,N=2 in a 16x16x4 tile) and t4
// is per-matrix, so matrix cores cannot help; the CDNA5 feature that matters
// is the async data-movement path. We stream x via GLOBAL_LOAD_ASYNC_TO_LDS
// (ASYNCcnt-pipelined, 4 stages, no VGPRs held by in-flight loads), compute
// ~20 packed-f32 VALU ops per thread, and emit non-temporal b128 stores.
// ============================================================================

typedef float v2f __attribute__((ext_vector_type(2)));
typedef float v4f __attribute__((ext_vector_type(4)));
typedef int v4i __attribute__((vector_size(16)));  // matches builtin pointee

#define GLOBAL_AS __attribute__((address_space(1)))
#define LDS_AS    __attribute__((address_space(3)))

#if defined(__HIP_DEVICE_COMPILE__) &&                                        \
    __has_builtin(__builtin_amdgcn_global_load_async_to_lds_b128) &&          \
    __has_builtin(__builtin_amdgcn_s_wait_asynccnt)
#define HAVE_ASYNC_BUILTINS 1
#else
#define HAVE_ASYNC_BUILTINS 0
#endif

// --- CDNA5 async global->LDS b128 copy (tracked by ASYNCcnt) ----------------
// Builtin signature (from hipcc diagnostic): param0 = int4 AS1* (global src),
// param1 = int4 AS3* (LDS dst), then imm offset, imm cpol.
__device__ __forceinline__ void async_copy_b128(const v4f* gsrc, v4f* lds_dst) {
#if defined(__HIP_DEVICE_COMPILE__)
#if HAVE_ASYNC_BUILTINS
  __builtin_amdgcn_global_load_async_to_lds_b128(
      (GLOBAL_AS v4i*)gsrc, (LDS_AS v4i*)lds_dst, 0, 0);
#else
  unsigned lds_off = (unsigned)(uintptr_t)(LDS_AS v4i*)lds_dst;
  asm volatile("global_load_async_to_lds_b128 %0, %1, off"
               :
               : "v"(lds_off), "v"(gsrc)
               : "memory");
#endif
#endif
}

template <int N>
__device__ __forceinline__ void wait_async() {
#if defined(__HIP_DEVICE_COMPILE__)
#if HAVE_ASYNC_BUILTINS
  __builtin_amdgcn_s_wait_asynccnt(N);
#else
  asm volatile("s_wait_asynccnt %0" : : "n"(N) : "memory");
#endif
#endif
}

// --- per-matrix chain: rows as float2, scalar*vector FMAs -------------------
__device__ __forceinline__ v4f eps_chain(v4f xv, v2f m0, v2f m1) {
  // t2 = x @ A0inv
  v2f t0 = xv.x * m0 + xv.y * m1;
  v2f t1 = xv.z * m0 + xv.w * m1;
  // t4 = t2 @ t2
  v2f q0 = t0.x * t0 + t0.y * t1;
  v2f q1 = t1.x * t0 + t1.y * t1;
  // u = I - t2 ; v = I + t4
  v2f u0 = {1.0f - t0.x, -t0.y};
  v2f u1 = {-t1.x, 1.0f - t1.y};
  v2f v0 = {1.0f + q0.x, q0.y};
  v2f v1 = {q1.x, 1.0f + q1.y};
  // ft = (A0inv @ u) @ v   (matches left-assoc evaluation in the reference)
  v2f g0 = m0.x * u0 + m0.y * u1;
  v2f g1 = m1.x * u0 + m1.y * u1;
  v2f f0 = g0.x * v0 + g0.y * v1;
  v2f f1 = g1.x * v0 + g1.y * v1;
  return (v4f){f0.x, f0.y, f1.x, f1.y};
}

constexpr int TPB    = 256;          // 8 waves (wave32)
constexpr int STAGES = 4;            // async pipeline depth
constexpr int TILE   = TPB * STAGES; // 1024 matrices / block, 16 KB LDS

__global__ __launch_bounds__(TPB) void eps_stream_kernel(
    const v4f* __restrict__ x, const float* __restrict__ A0inv,
    v4f* __restrict__ out) {
  __shared__ v4f smem[TILE];
  const int tid = threadIdx.x;
  const size_t base = (size_t)blockIdx.x * TILE;

  // Issue all 4 stage loads; lane t stage s -> matrix base + s*TPB + t
  // (16B/lane, fully coalesced 512B per wave-instruction).
  async_copy_b128(x + base + 0 * TPB + tid, smem + 0 * TPB + tid);
  async_copy_b128(x + base + 1 * TPB + tid, smem + 1 * TPB + tid);
  async_copy_b128(x + base + 2 * TPB + tid, smem + 2 * TPB + tid);
  async_copy_b128(x + base + 3 * TPB + tid, smem + 3 * TPB + tid);

  const v2f m0 = {A0inv[0], A0inv[1]};
  const v2f m1 = {A0inv[2], A0inv[3]};

  // Async loads complete in order -> drain stages as ASYNCcnt falls.
  // Each lane reads only its own LDS slot: no barrier needed.
  wait_async<3>();
  __builtin_nontemporal_store(eps_chain(smem[0 * TPB + tid], m0, m1),
                              out + base + 0 * TPB + tid);
  wait_async<2>();
  __builtin_nontemporal_store(eps_chain(smem[1 * TPB + tid], m0, m1),
                              out + base + 1 * TPB + tid);
  wait_async<1>();
  __builtin_nontemporal_store(eps_chain(smem[2 * TPB + tid], m0, m1),
                              out + base + 2 * TPB + tid);
  wait_async<0>();
  __builtin_nontemporal_store(eps_chain(smem[3 * TPB + tid], m0, m1),
                              out + base + 3 * TPB + tid);
}

// Remainder path (B % 1024 != 0): plain NT b128 load/store.
__global__ __launch_bounds__(TPB) void eps_tail_kernel(
    const v4f* __restrict__ x, const float* __restrict__ A0inv,
    v4f* __restrict__ out, long long start, long long n) {
  long long i = start + (long long)blockIdx.x * TPB + threadIdx.x;
  if (i < n) {
    v2f m0 = {A0inv[0], A0inv[1]};
    v2f m1 = {A0inv[2], A0inv[3]};
    v4f xv = __builtin_nontemporal_load(x + i);
    __builtin_nontemporal_store(eps_chain(xv, m0, m1), out + i);
  }
}

extern "C" void kernel_launch(void* const* d_in, const int* in_sizes, int n_in,
                              void* d_out, int out_size, void* d_ws,
                              size_t ws_size, hipStream_t stream) {
  const v4f* x = (const v4f*)d_in[0];          // [B,2,2] fp32 -> v4f per matrix
  const float* A0inv = (const float*)d_in[1];  // [2,2] fp32
  v4f* out = (v4f*)d_out;                      // [B,2,2] fp32

  const long long nmat = (long long)in_sizes[0] / 4;  // 8,388,608
  const long long nfull = nmat / TILE;                // 8192 blocks
  if (nfull > 0) {
    eps_stream_kernel<<<dim3((unsigned)nfull), dim3(TPB), 0, stream>>>(
        x, A0inv, out);
  }
  const long long rem = nmat - nfull * TILE;          // 0 for B = 8.4M
  if (rem > 0) {
    eps_tail_kernel<<<dim3((unsigned)((rem + TPB - 1) / TPB)), dim3(TPB), 0,
                      stream>>>(x, A0inv, out, nfull * TILE, nmat);
  }
}